// ConvAttention_14585708937724
// MI455X (gfx1250) — compile-verified
//
#include <hip/hip_runtime.h>
#include <hip/hip_bf16.h>
#include <math.h>

typedef float v2f __attribute__((ext_vector_type(2)));
typedef float v8f __attribute__((ext_vector_type(8)));

#define B_   8
#define T1_  768
#define T2_  192
#define NS_  15
#define NT_  256
#define NT2_ 512
#define NA_  64

// ---------------------------------------------------------------------------
// Kernel 1: key projection.  One 16-row tile of (b,t2) per workgroup.
//   dwconv(k=3,pad=1) -> LDS, WMMA fp32 GEMM 256->512 (+bias,SiLU,k2 dw scale)
//   -> LDS, WMMA fp32 GEMM 512->64 (+bias) -> ke, knorm.
// ---------------------------------------------------------------------------
__global__ __launch_bounds__(256)
void key_proj_kernel(const float* __restrict__ keys,
                     const float* __restrict__ k1_dw, const float* __restrict__ k1_db,
                     const float* __restrict__ k1_pw, const float* __restrict__ k1_pb,
                     const float* __restrict__ k2_dw, const float* __restrict__ k2_db,
                     const float* __restrict__ k2_pw, const float* __restrict__ k2_pb,
                     float* __restrict__ ke, float* __restrict__ knorm)
{
    // padded row strides (260, 516): stride % 64 == 4 -> conflict-free column reads
    __shared__ float At[16][NT_ + 4];
    __shared__ float Ht[16][NT2_ + 4];

    const int tile = blockIdx.x;               // 0..95
    const int b    = tile / (T2_ / 16);
    const int t0   = (tile % (T2_ / 16)) * 16;
    const int tid  = threadIdx.x;
    const int lane = tid & 31;
    const int wave = tid >> 5;                 // 0..7
    const int mrow = lane & 15;
    const int kh   = lane >> 4;                // lane half selects K pair

    // Phase A: depthwise conv k=3 pad=1; thread tid owns channel c=tid
    {
        const int c = tid;
        const float w0 = k1_dw[0 * NT_ + c];
        const float w1 = k1_dw[1 * NT_ + c];
        const float w2 = k1_dw[2 * NT_ + c];
        const float bi = k1_db[c];
        for (int m = 0; m < 16; ++m) {
            const int t = t0 + m;
            const float xm1 = (t > 0)        ? keys[((size_t)(b * T2_ + t - 1)) * NT_ + c] : 0.f;
            const float x0  =                  keys[((size_t)(b * T2_ + t    )) * NT_ + c];
            const float xp1 = (t < T2_ - 1)  ? keys[((size_t)(b * T2_ + t + 1)) * NT_ + c] : 0.f;
            At[m][c] = w0 * xm1 + w1 * x0 + w2 * xp1 + bi;
        }
    }
    __syncthreads();

    // Phase B: 16xNT_ @ NT_xNT2_ via V_WMMA_F32_16X16X4_F32; each wave does 4 N-tiles
    for (int nt = wave * 4; nt < wave * 4 + 4; ++nt) {
        const int n0 = nt * 16;
        v8f acc = {};
        for (int ks = 0; ks < NT_ / 4; ++ks) {
            const int k = ks * 4 + kh * 2;
            v2f a, bf;
            a[0]  = At[mrow][k];
            a[1]  = At[mrow][k + 1];
            bf[0] = k1_pw[(size_t)(k    ) * NT2_ + n0 + mrow];
            bf[1] = k1_pw[(size_t)(k + 1) * NT2_ + n0 + mrow];
            acc = __builtin_amdgcn_wmma_f32_16x16x4_f32(false, a, false, bf,
                                                        (short)0, acc, false, false);
        }
        const int   n   = n0 + mrow;
        const float pb  = k1_pb[n];
        const float dw2 = k2_dw[n];
        const float db2 = k2_db[n];
        for (int v = 0; v < 8; ++v) {
            const int m = v + 8 * kh;
            float u = acc[v] + pb;
            float s = u / (1.f + __expf(-u));      // SiLU
            Ht[m][n] = s * dw2 + db2;              // k=1 depthwise of second sepconv
        }
    }
    __syncthreads();

    // Phase C: 16xNT2_ @ NT2_xNA_ ; waves 0..3 take one 16-col tile each
    if (wave < 4) {
        const int n0 = wave * 16;
        v8f acc = {};
        for (int ks = 0; ks < NT2_ / 4; ++ks) {
            const int k = ks * 4 + kh * 2;
            v2f a, bf;
            a[0]  = Ht[mrow][k];
            a[1]  = Ht[mrow][k + 1];
            bf[0] = k2_pw[(size_t)(k    ) * NA_ + n0 + mrow];
            bf[1] = k2_pw[(size_t)(k + 1) * NA_ + n0 + mrow];
            acc = __builtin_amdgcn_wmma_f32_16x16x4_f32(false, a, false, bf,
                                                        (short)0, acc, false, false);
        }
        const int   n  = n0 + mrow;
        const float pb = k2_pb[n];
        for (int v = 0; v < 8; ++v) {
            const int m = v + 8 * kh;
            const float val = acc[v] + pb;
            ke[((size_t)(b * T2_ + t0 + m)) * NA_ + n] = val;
            At[m][n] = val;                        // reuse LDS for norm pass
        }
    }
    __syncthreads();

    if (tid < 16) {
        float s = 0.f;
        for (int c = 0; c < NA_; ++c) { const float x = At[tid][c]; s += x * x; }
        knorm[b * T2_ + t0 + tid] = s;
    }
}

// ---------------------------------------------------------------------------
// Kernel 2: query projection (15->30->15->64), one thread per (b,t1) row.
// ---------------------------------------------------------------------------
__global__ __launch_bounds__(256)
void query_proj_kernel(const float* __restrict__ q,
                       const float* __restrict__ q1_dw, const float* __restrict__ q1_db,
                       const float* __restrict__ q1_pw, const float* __restrict__ q1_pb,
                       const float* __restrict__ q2_dw, const float* __restrict__ q2_db,
                       const float* __restrict__ q2_pw, const float* __restrict__ q2_pb,
                       const float* __restrict__ q3_dw, const float* __restrict__ q3_db,
                       const float* __restrict__ q3_pw, const float* __restrict__ q3_pb,
                       float* __restrict__ qe, float* __restrict__ qnorm)
{
    const int r = blockIdx.x * blockDim.x + threadIdx.x;
    if (r >= B_ * T1_) return;
    const int b = r / T1_;
    const int t = r % T1_;

    float x[NS_];
    for (int c = 0; c < NS_; ++c) {
        const float xm1 = (t > 0)        ? q[((size_t)(b * T1_ + t - 1)) * NS_ + c] : 0.f;
        const float x0  =                  q[((size_t)(b * T1_ + t    )) * NS_ + c];
        const float xp1 = (t < T1_ - 1)  ? q[((size_t)(b * T1_ + t + 1)) * NS_ + c] : 0.f;
        x[c] = q1_dw[c] * xm1 + q1_dw[NS_ + c] * x0 + q1_dw[2 * NS_ + c] * xp1 + q1_db[c];
    }
    float u[2 * NS_];
    for (int n = 0; n < 2 * NS_; ++n) {
        float s = q1_pb[n];
        for (int c = 0; c < NS_; ++c) s += x[c] * q1_pw[c * (2 * NS_) + n];
        const float sl = s / (1.f + __expf(-s));   // SiLU
        u[n] = sl * q2_dw[n] + q2_db[n];           // k=1 dw of sepconv 2
    }
    float w[NS_];
    for (int n = 0; n < NS_; ++n) {
        float s = q2_pb[n];
        for (int c = 0; c < 2 * NS_; ++c) s += u[c] * q2_pw[c * NS_ + n];
        const float sl = s / (1.f + __expf(-s));   // SiLU
        w[n] = sl * q3_dw[n] + q3_db[n];           // k=1 dw of sepconv 3
    }
    float nrm = 0.f;
    for (int n = 0; n < NA_; ++n) {
        float s = q3_pb[n];
        for (int c = 0; c < NS_; ++c) s += w[c] * q3_pw[c * NA_ + n];
        qe[(size_t)r * NA_ + n] = s;
        nrm += s * s;
    }
    qnorm[r] = nrm;
}

// ---------------------------------------------------------------------------
// Kernel 3: fused QK^T (WMMA fp32) + log-softmax + prior + masked softmax.
// One wave per 16-row tile; 12 accumulator tiles cover all 192 columns.
// ---------------------------------------------------------------------------
__global__ __launch_bounds__(32)
void attn_kernel(const float* __restrict__ qe, const float* __restrict__ qnorm,
                 const float* __restrict__ ke, const float* __restrict__ knorm,
                 const float* __restrict__ prior, const unsigned char* __restrict__ mask,
                 float* __restrict__ attn, float* __restrict__ logprob)
{
    const int blk  = blockIdx.x;               // 0..383
    const int b    = blk / (T1_ / 16);
    const int i0   = (blk % (T1_ / 16)) * 16;
    const int lane = threadIdx.x;              // 0..31
    const int nn   = lane & 15;                // column within tile / row of fragment
    const int h    = lane >> 4;                // lane half

    v8f acc[12];
    for (int t = 0; t < 12; ++t) acc[t] = (v8f){0, 0, 0, 0, 0, 0, 0, 0};

    const float* qrow  = qe + ((size_t)(b * T1_ + i0)) * NA_;
    const float* kbase = ke + (size_t)b * T2_ * NA_;
    for (int ks = 0; ks < NA_ / 4; ++ks) {
        const int k = ks * 4 + h * 2;
        v2f a;
        a[0] = qrow[nn * NA_ + k];
        a[1] = qrow[nn * NA_ + k + 1];
        for (int t = 0; t < 12; ++t) {
            v2f bf;                            // B[k][n] = ke[n][k] (transposed access)
            bf[0] = kbase[(size_t)(t * 16 + nn) * NA_ + k];
            bf[1] = kbase[(size_t)(t * 16 + nn) * NA_ + k + 1];
            acc[t] = __builtin_amdgcn_wmma_f32_16x16x4_f32(false, a, false, bf,
                                                           (short)0, acc[t], false, false);
        }
    }

    // logits: -5e-4 * (|q|^2 + |k|^2 - 2 q.k)
    float qn[8];
    for (int v = 0; v < 8; ++v) qn[v] = qnorm[b * T1_ + i0 + v + 8 * h];
    float kn[12];
    for (int t = 0; t < 12; ++t) kn[t] = knorm[b * T2_ + t * 16 + nn];
    for (int t = 0; t < 12; ++t)
        for (int v = 0; v < 8; ++v)
            acc[t][v] = -5e-4f * (qn[v] + kn[t] - 2.f * acc[t][v]);

    // pass 1: log-softmax over the 192 columns (per row v; xor masks 1..8 stay
    // inside each 16-lane half, which is exactly one set of rows under wave32)
    float lse[8];
    for (int v = 0; v < 8; ++v) {
        float m = -INFINITY;
        for (int t = 0; t < 12; ++t) m = fmaxf(m, acc[t][v]);
        for (int o = 1; o < 16; o <<= 1) m = fmaxf(m, __shfl_xor(m, o, 32));
        float s = 0.f;
        for (int t = 0; t < 12; ++t) s += expf(acc[t][v] - m);
        for (int o = 1; o < 16; o <<= 1) s += __shfl_xor(s, o, 32);
        lse[v] = m + logf(s);
    }

    const float* prow  = prior  + ((size_t)(b * T1_ + i0)) * T2_;
    float*       lpout = logprob + ((size_t)(b * T1_ + i0)) * T2_;
    for (int t = 0; t < 12; ++t) {
        for (int v = 0; v < 8; ++v) {
            const int m = v + 8 * h;
            const int j = t * 16 + nn;
            const float lp = acc[t][v] - lse[v] + logf(prow[m * T2_ + j] + 1e-8f);
            lpout[m * T2_ + j] = lp;
            acc[t][v] = lp;
        }
    }

    // pass 2: masked softmax
    bool mk[12];
    for (int t = 0; t < 12; ++t) mk[t] = mask[b * T2_ + t * 16 + nn] != 0;
    float* aout = attn + ((size_t)(b * T1_ + i0)) * T2_;
    for (int v = 0; v < 8; ++v) {
        float m = -INFINITY;
        for (int t = 0; t < 12; ++t) m = fmaxf(m, mk[t] ? -INFINITY : acc[t][v]);
        for (int o = 1; o < 16; o <<= 1) m = fmaxf(m, __shfl_xor(m, o, 32));
        float s = 0.f;
        for (int t = 0; t < 12; ++t) s += mk[t] ? 0.f : expf(acc[t][v] - m);
        for (int o = 1; o < 16; o <<= 1) s += __shfl_xor(s, o, 32);
        const float inv = 1.f / s;
        for (int t = 0; t < 12; ++t) {
            const int row = v + 8 * h;
            const int j   = t * 16 + nn;
            aout[row * T2_ + j] = mk[t] ? 0.f : expf(acc[t][v] - m) * inv;
        }
    }
}

// ---------------------------------------------------------------------------
extern "C" void kernel_launch(void* const* d_in, const int* in_sizes, int n_in,
                              void* d_out, int out_size, void* d_ws, size_t ws_size,
                              hipStream_t stream)
{
    (void)in_sizes; (void)n_in; (void)out_size; (void)ws_size;

    const float*         queries = (const float*)d_in[0];
    const float*         keys    = (const float*)d_in[1];
    const unsigned char* mask    = (const unsigned char*)d_in[2];  // jnp bool = 1 byte
    const float*         prior   = (const float*)d_in[3];
    const float* k1_dw = (const float*)d_in[4];
    const float* k1_db = (const float*)d_in[5];
    const float* k1_pw = (const float*)d_in[6];
    const float* k1_pb = (const float*)d_in[7];
    const float* k2_dw = (const float*)d_in[8];
    const float* k2_db = (const float*)d_in[9];
    const float* k2_pw = (const float*)d_in[10];
    const float* k2_pb = (const float*)d_in[11];
    const float* q1_dw = (const float*)d_in[12];
    const float* q1_db = (const float*)d_in[13];
    const float* q1_pw = (const float*)d_in[14];
    const float* q1_pb = (const float*)d_in[15];
    const float* q2_dw = (const float*)d_in[16];
    const float* q2_db = (const float*)d_in[17];
    const float* q2_pw = (const float*)d_in[18];
    const float* q2_pb = (const float*)d_in[19];
    const float* q3_dw = (const float*)d_in[20];
    const float* q3_db = (const float*)d_in[21];
    const float* q3_pw = (const float*)d_in[22];
    const float* q3_pb = (const float*)d_in[23];

    float* out     = (float*)d_out;
    float* attn    = out;                                   // (B,1,T1,T2)
    float* logprob = out + (size_t)B_ * T1_ * T2_;          // (B,1,T1,T2)

    float* ws    = (float*)d_ws;
    float* ke    = ws;                                      // B*T2*NA
    float* knorm = ke + (size_t)B_ * T2_ * NA_;             // B*T2
    float* qe    = knorm + (size_t)B_ * T2_;                // B*T1*NA
    float* qnorm = qe + (size_t)B_ * T1_ * NA_;             // B*T1

    key_proj_kernel<<<B_ * (T2_ / 16), 256, 0, stream>>>(
        keys, k1_dw, k1_db, k1_pw, k1_pb, k2_dw, k2_db, k2_pw, k2_pb, ke, knorm);

    query_proj_kernel<<<(B_ * T1_) / 256, 256, 0, stream>>>(
        queries, q1_dw, q1_db, q1_pw, q1_pb, q2_dw, q2_db, q2_pw, q2_pb,
        q3_dw, q3_db, q3_pw, q3_pb, qe, qnorm);

    attn_kernel<<<B_ * (T1_ / 16), 32, 0, stream>>>(
        qe, qnorm, ke, knorm, prior, mask, attn, logprob);
}